// HeteroRGCNLayer_76227079569906
// MI455X (gfx1250) — compile-verified
//
#include <hip/hip_runtime.h>

typedef float v2f __attribute__((ext_vector_type(2)));
typedef float v8f __attribute__((ext_vector_type(8)));

#define N_USER 100000
#define N_ITEM 100000
#define NNODE  100000            // both ntypes happen to be 100k
#define NEDGE  1600000
#define DDIM   64
#define NU64   (N_USER * DDIM)   // 6,400,000 floats
#define SCAN_B 256
#define NB_SCAN ((NNODE + SCAN_B - 1) / SCAN_B)   // 391
#define NB_PAD 512

// ===========================================================================
// Shared: zero kernels
// ===========================================================================
__global__ void zero_f32(float* __restrict__ p, int n) {
    int i = blockIdx.x * blockDim.x + threadIdx.x;
    if (i < n) p[i] = 0.0f;
}
__global__ void zero_i32(int* __restrict__ p, int n) {
    int i = blockIdx.x * blockDim.x + threadIdx.x;
    if (i < n) p[i] = 0;
}

// ===========================================================================
// Projection: Y[m][n] = sum_k X[m][k] * W[n][k] + b[n]   (x @ W.T + b)
// One wave32 -> 16x64 output tile via V_WMMA_F32_16X16X4_F32 (16 K-steps x 4
// N-tiles). 100000 rows = 6250 exact 16-row tiles.
// ===========================================================================
__global__ void project_wmma(const float* __restrict__ X,
                             const float* __restrict__ W,
                             const float* __restrict__ bias,
                             float* __restrict__ Y,
                             int nrows) {
    int wave = (int)((blockIdx.x * blockDim.x + threadIdx.x) >> 5);
    int lane = threadIdx.x & 31;
    int row0 = wave * 16;
    if (row0 >= nrows) return;

    int half = lane >> 4;      // 0: K={0,1}, 1: K={2,3} within a K-step
    int l15  = lane & 15;
    int m    = row0 + l15;     // A-matrix row for this lane
    int koff = half * 2;

    v8f acc[4];
#pragma unroll
    for (int t = 0; t < 4; ++t) acc[t] = (v8f){0,0,0,0,0,0,0,0};

    for (int kt = 0; kt < DDIM; kt += 4) {
        v2f a;
        a.x = X[m * DDIM + kt + koff];
        a.y = X[m * DDIM + kt + koff + 1];
#pragma unroll
        for (int t = 0; t < 4; ++t) {
            int n = t * 16 + l15;                 // output feature
            v2f b;
            b.x = W[n * DDIM + kt + koff];        // B[k][n] = W[n][k]
            b.y = W[n * DDIM + kt + koff + 1];
            acc[t] = __builtin_amdgcn_wmma_f32_16x16x4_f32(
                false, a, false, b, (short)0, acc[t], false, false);
        }
    }

#pragma unroll
    for (int t = 0; t < 4; ++t) {
        int n  = t * 16 + l15;
        float bv = bias[n];
#pragma unroll
        for (int v = 0; v < 8; ++v) {
            int mrow = row0 + v + half * 8;
            Y[mrow * DDIM + n] = acc[t][v] + bv;
        }
    }
}

// ===========================================================================
// PULL PATH (CSR gather): no f32 atomics in the hot loop.
// ===========================================================================
__global__ void degree_count_i(const int* __restrict__ dst,
                               int* __restrict__ cnt, int nedges) {
    int i = blockIdx.x * blockDim.x + threadIdx.x;
    if (i < nedges) atomicAdd(&cnt[dst[i]], 1);
}

// Per-block exclusive scan of counts; emits per-block totals.
__global__ void scan_local(const int* __restrict__ cnt,
                           int* __restrict__ off,
                           int* __restrict__ bsum, int n) {
    __shared__ int sm[SCAN_B];
    int t = threadIdx.x;
    int i = blockIdx.x * SCAN_B + t;
    int x = (i < n) ? cnt[i] : 0;
    sm[t] = x;
    __syncthreads();
    for (int o = 1; o < SCAN_B; o <<= 1) {
        int v = (t >= o) ? sm[t - o] : 0;
        __syncthreads();
        sm[t] += v;
        __syncthreads();
    }
    if (i < n) off[i] = sm[t] - x;          // exclusive
    if (t == SCAN_B - 1) bsum[blockIdx.x] = sm[t];
}

// Single-block exclusive scan of the (<=512) block totals.
__global__ void scan_bsums(int* __restrict__ bsum, int nb) {
    __shared__ int sm[NB_PAD];
    int t = threadIdx.x;
    int x = (t < nb) ? bsum[t] : 0;
    sm[t] = x;
    __syncthreads();
    for (int o = 1; o < NB_PAD; o <<= 1) {
        int v = (t >= o) ? sm[t - o] : 0;
        __syncthreads();
        sm[t] += v;
        __syncthreads();
    }
    if (t < nb) bsum[t] = sm[t] - x;        // exclusive
}

// Add block bases; also seed the binning cursor (= final CSR offset).
__global__ void scan_add(int* __restrict__ off,
                         const int* __restrict__ bsum,
                         int* __restrict__ cursor, int n) {
    int i = blockIdx.x * blockDim.x + threadIdx.x;
    if (i < n) {
        int o = off[i] + bsum[i / SCAN_B];
        off[i] = o;
        cursor[i] = o;
    }
}

// Scatter edge source-ids into CSR slots (returning u32 atomics, cheap).
__global__ void bin_edges(const int* __restrict__ src,
                          const int* __restrict__ dst,
                          int* __restrict__ cursor,
                          int* __restrict__ esrc, int nedges) {
    int i = blockIdx.x * blockDim.x + threadIdx.x;
    if (i < nedges) {
        int slot = atomicAdd(&cursor[dst[i]], 1);
        esrc[slot] = src[i];
    }
}

// One wave per destination node. Lane l owns features {2l, 2l+1}. Loop over
// the node's in-edges with float2 gathers of the projected source row, mean,
// then a single vector store (or load-add-store for the 2nd user etype).
__global__ void gather_mean(const float* __restrict__ Wh,
                            const int* __restrict__ esrc,
                            const int* __restrict__ off,
                            const int* __restrict__ cnt,
                            float* __restrict__ out,
                            int nnodes, int accumulate) {
    int node = (int)((blockIdx.x * blockDim.x + threadIdx.x) >> 5);
    int lane = threadIdx.x & 31;
    if (node >= nnodes) return;

    int start = off[node];
    int deg   = cnt[node];

    float2 acc = make_float2(0.0f, 0.0f);
    for (int j = 0; j < deg; ++j) {
        int s = esrc[start + j];   // same address all lanes -> broadcast hit
        const float2 v = *(const float2*)(Wh + (size_t)s * DDIM + 2 * lane);
        acc.x += v.x;
        acc.y += v.y;
    }
    float inv = 1.0f / fmaxf((float)deg, 1.0f);
    acc.x *= inv;
    acc.y *= inv;

    float* o = out + (size_t)node * DDIM + 2 * lane;
    if (accumulate) {
        float2 p = *(const float2*)o;
        acc.x += p.x;
        acc.y += p.y;
    }
    *(float2*)o = acc;
}

// ===========================================================================
// PUSH FALLBACK (if workspace too small for CSR): scaled scatter-atomics.
// ===========================================================================
__global__ void degree_count_f(const int* __restrict__ dst,
                               float* __restrict__ cnt, int nedges) {
    int i = blockIdx.x * blockDim.x + threadIdx.x;
    if (i < nedges) atomicAdd(&cnt[dst[i]], 1.0f);
}
__global__ void recip_clamp(float* __restrict__ c, int n) {
    int i = blockIdx.x * blockDim.x + threadIdx.x;
    if (i < n) c[i] = 1.0f / fmaxf(c[i], 1.0f);
}
__global__ void scatter_mean(const float* __restrict__ Wh,
                             const int* __restrict__ src,
                             const int* __restrict__ dst,
                             const float* __restrict__ inv_cnt,
                             float* __restrict__ out, int nedges) {
    long long idx = (long long)blockIdx.x * blockDim.x + threadIdx.x;
    int e = (int)(idx >> 6);
    int d = (int)(idx & 63);
    if (e >= nedges) return;
    int s = src[e];
    int t = dst[e];
    atomicAdd(&out[t * DDIM + d], Wh[s * DDIM + d] * inv_cnt[t]);
}

// ===========================================================================
extern "C" void kernel_launch(void* const* d_in, const int* in_sizes, int n_in,
                              void* d_out, int out_size, void* d_ws, size_t ws_size,
                              hipStream_t stream) {
    const float* feat_user = (const float*)d_in[0];
    const float* feat_item = (const float*)d_in[1];
    const float* W_follows = (const float*)d_in[2];
    const float* b_follows = (const float*)d_in[3];
    const float* W_rates   = (const float*)d_in[4];
    const float* b_rates   = (const float*)d_in[5];
    const float* W_ratedby = (const float*)d_in[6];
    const float* b_ratedby = (const float*)d_in[7];
    const int* src_follows = (const int*)d_in[8];
    const int* dst_follows = (const int*)d_in[9];
    const int* src_rates   = (const int*)d_in[10];
    const int* dst_rates   = (const int*)d_in[11];
    const int* src_ratedby = (const int*)d_in[12];
    const int* dst_ratedby = (const int*)d_in[13];

    float* out_user = (float*)d_out;           // [N_USER, 64]
    float* out_item = out_user + NU64;         // [N_ITEM, 64]

    // ---- workspace layout -------------------------------------------------
    float* ws    = (float*)d_ws;
    float* Wh_f  = ws;
    float* Wh_r  = ws + (size_t)NU64;
    float* Wh_rb = ws + (size_t)2 * NU64;

    const size_t push_bytes = ((size_t)3 * NU64 + 3 * NNODE) * 4;
    const size_t pull_bytes = ((size_t)3 * NU64 + 9 * NNODE + 3 * NB_PAD
                               + (size_t)3 * NEDGE) * 4;
    const bool use_pull = (ws_size >= pull_bytes);

    // ---- projections (shared by both paths) ------------------------------
    {
        int waves  = N_USER / 16;              // 6250 exact
        int blocks = (waves + 7) / 8;          // 8 waves / 256-thread block
        project_wmma<<<blocks, 256, 0, stream>>>(feat_user, W_follows, b_follows, Wh_f,  N_USER);
        project_wmma<<<blocks, 256, 0, stream>>>(feat_user, W_rates,   b_rates,   Wh_r,  N_USER);
        project_wmma<<<blocks, 256, 0, stream>>>(feat_item, W_ratedby, b_ratedby, Wh_rb, N_ITEM);
    }

    const int eb = (NEDGE + 255) / 256;        // 6250 blocks for edge sweeps

    if (use_pull) {
        // layout after Wh: cnt[3N] | off[3N] | cur[3N] | bsum[3*512] | esrc[3E]
        int* cnt  = (int*)(ws + (size_t)3 * NU64);
        int* off  = cnt + 3 * NNODE;
        int* cur  = off + 3 * NNODE;
        int* bsum = cur + 3 * NNODE;
        int* esrc = bsum + 3 * NB_PAD;

        const int* dsts[3] = { dst_follows, dst_ratedby, dst_rates };
        const int* srcs[3] = { src_follows, src_ratedby, src_rates };

        // 1) degree histograms (int atomics)
        zero_i32<<<(3 * NNODE + 255) / 256, 256, 0, stream>>>(cnt, 3 * NNODE);
        for (int k = 0; k < 3; ++k)
            degree_count_i<<<eb, 256, 0, stream>>>(dsts[k], cnt + k * NNODE, NEDGE);

        // 2) exclusive scans -> CSR offsets + binning cursors
        for (int k = 0; k < 3; ++k) {
            scan_local<<<NB_SCAN, SCAN_B, 0, stream>>>(cnt + k * NNODE,
                                                       off + k * NNODE,
                                                       bsum + k * NB_PAD, NNODE);
            scan_bsums<<<1, NB_PAD, 0, stream>>>(bsum + k * NB_PAD, NB_SCAN);
            scan_add<<<(NNODE + 255) / 256, 256, 0, stream>>>(off + k * NNODE,
                                                              bsum + k * NB_PAD,
                                                              cur + k * NNODE, NNODE);
        }

        // 3) bin edge source-ids into CSR
        for (int k = 0; k < 3; ++k)
            bin_edges<<<eb, 256, 0, stream>>>(srcs[k], dsts[k],
                                              cur + k * NNODE,
                                              esrc + (size_t)k * NEDGE, NEDGE);

        // 4) pull-gather: one wave per node, no f32 atomics, no out zeroing
        int gblocks = (NNODE * 32 + 255) / 256;   // 8 waves / block
        gather_mean<<<gblocks, 256, 0, stream>>>(Wh_f,  esrc + 0ll * NEDGE,
                                                 off + 0 * NNODE, cnt + 0 * NNODE,
                                                 out_user, NNODE, 0);
        gather_mean<<<gblocks, 256, 0, stream>>>(Wh_rb, esrc + 1ll * NEDGE,
                                                 off + 1 * NNODE, cnt + 1 * NNODE,
                                                 out_user, NNODE, 1);
        gather_mean<<<gblocks, 256, 0, stream>>>(Wh_r,  esrc + 2ll * NEDGE,
                                                 off + 2 * NNODE, cnt + 2 * NNODE,
                                                 out_item, NNODE, 0);
    } else {
        // -------- push fallback: scaled scatter-add with f32 atomics -------
        float* cnt_f  = ws + (size_t)3 * NU64;
        float* cnt_rb = cnt_f + NNODE;
        float* cnt_r  = cnt_rb + NNODE;
        const int n_cnt = 3 * NNODE;

        zero_f32<<<(2 * NU64 + 255) / 256, 256, 0, stream>>>((float*)d_out, 2 * NU64);
        zero_f32<<<(n_cnt + 255) / 256, 256, 0, stream>>>(cnt_f, n_cnt);

        degree_count_f<<<eb, 256, 0, stream>>>(dst_follows, cnt_f,  NEDGE);
        degree_count_f<<<eb, 256, 0, stream>>>(dst_ratedby, cnt_rb, NEDGE);
        degree_count_f<<<eb, 256, 0, stream>>>(dst_rates,   cnt_r,  NEDGE);
        recip_clamp<<<(n_cnt + 255) / 256, 256, 0, stream>>>(cnt_f, n_cnt);

        long long total = (long long)NEDGE * DDIM;
        int blocks = (int)((total + 255) / 256);
        scatter_mean<<<blocks, 256, 0, stream>>>(Wh_f,  src_follows, dst_follows, cnt_f,  out_user, NEDGE);
        scatter_mean<<<blocks, 256, 0, stream>>>(Wh_rb, src_ratedby, dst_ratedby, cnt_rb, out_user, NEDGE);
        scatter_mean<<<blocks, 256, 0, stream>>>(Wh_r,  src_rates,   dst_rates,   cnt_r,  out_item, NEDGE);
    }
}